// GTEProgramClassification_27986006900812
// MI455X (gfx1250) — compile-verified
//
#include <hip/hip_runtime.h>
#include <math.h>

#define D 128
#define LSTRIDE 132            // padded LDS row stride for activations (floats)
#define WSTRIDE2 288           // paired-K weight row stride (floats); 288%64==32 -> conflict-free
#define LCH 5
#define NCHILD 4
#define N_DST 100000
#define NCLS 104
#define NCLS_PAD 112
#define BLOCK_ROWS 128
#define THREADS 256

typedef float v2f __attribute__((ext_vector_type(2)));
typedef float v8f __attribute__((ext_vector_type(8)));

__device__ __forceinline__ float sigmoid_f(float x) {
    return 1.0f / (1.0f + __expf(-x));
}

// Stage a 128x128 row-major fp32 weight into LDS in K-paired layout:
//   Wp[k2*WSTRIDE2 + 2*n + {0,1}] = { W[2*k2][n], W[2*k2+1][n] }
// so a WMMA B-fragment is a single aligned 8-byte LDS load per lane.
__device__ __forceinline__ void load_w128_paired(const float* __restrict__ W,
                                                 float* Wp, int tid) {
    #pragma unroll
    for (int it = 0; it < 8; ++it) {
        int idx = tid + it * THREADS;          // 0..2047: (k2, float4-col)
        int k2  = idx >> 5;                    // 0..63
        int c4  = (idx & 31) << 2;             // 0,4,...,124
        float4 v0 = *(const float4*)(W + (2 * k2)     * D + c4);
        float4 v1 = *(const float4*)(W + (2 * k2 + 1) * D + c4);
        v2f* w = (v2f*)(Wp + k2 * WSTRIDE2 + 2 * c4);
        v2f p0 = {v0.x, v1.x}; v2f p1 = {v0.y, v1.y};
        v2f p2 = {v0.z, v1.z}; v2f p3 = {v0.w, v1.w};
        w[0] = p0; w[1] = p1; w[2] = p2; w[3] = p3;
    }
}

// One wave computes a 16-row strip (rows m0..m0+15) of S(128x128) @ W(128 x NT*16)
// via chained V_WMMA_F32_16X16X4_F32.
// A-fragment (16x4 f32): lane l holds S[m0+(l&15)][k+2*(l>>4)] , [..+1]  (one b64 load).
// B-fragment (4x16 f32): lane l holds W[k+2*(l>>4)][n] , W[k+2*(l>>4)+1][n] (one b64 load).
template <int NT>
__device__ __forceinline__ void gemm_strip(const float* __restrict__ Sb,
                                           const float* __restrict__ Wp,
                                           int m0, int ln16, int hi, v8f* acc) {
    #pragma unroll
    for (int t = 0; t < NT; ++t) {
        v8f z = {0.f, 0.f, 0.f, 0.f, 0.f, 0.f, 0.f, 0.f};
        acc[t] = z;
    }
    const float* arow = Sb + (m0 + ln16) * LSTRIDE;
    for (int k = 0; k < D; k += 4) {
        int kk = k + 2 * hi;                        // even
        v2f a = *(const v2f*)(arow + kk);           // ds_load_b64
        const float* brow = Wp + (kk >> 1) * WSTRIDE2 + 2 * ln16;
        #pragma unroll
        for (int t = 0; t < NT; ++t) {
            v2f b = *(const v2f*)(brow + 32 * t);   // ds_load_b64
            acc[t] = __builtin_amdgcn_wmma_f32_16x16x4_f32(
                false, a, false, b, 0, acc[t], false, false);
        }
    }
}

__global__ __launch_bounds__(THREADS)
void treelstm_cls_kernel(const int* __restrict__ token_ids,
                         const int* __restrict__ child_idx,
                         const float* __restrict__ emb,
                         const float* __restrict__ W_ih, const float* __restrict__ b_ih,
                         const float* __restrict__ W_oh, const float* __restrict__ b_oh,
                         const float* __restrict__ W_uh, const float* __restrict__ b_uh,
                         const float* __restrict__ ln_g, const float* __restrict__ ln_b,
                         const float* __restrict__ W_fc, const float* __restrict__ b_fc,
                         float* __restrict__ out) {
    extern __shared__ float smem[];
    float* S  = smem;                            // 128 x 132 activations (later h_norm)
    float* Wp = smem + BLOCK_ROWS * LSTRIDE;     // 64 x 288 paired-K weight tile

    const int tid  = threadIdx.x;
    const int lane = tid & 31;
    const int wave = tid >> 5;
    const int ln16 = lane & 15;
    const int hi   = lane >> 4;                  // wave32 half
    const int m0   = wave * 16;                  // this wave's row strip
    const int row0 = blockIdx.x * BLOCK_ROWS;

    // ---- Phase 1: gather 4 child embeddings per node, sum into S ----
    {
        const int r8 = tid >> 5;                 // row within 8-row iteration
        const int c4 = (tid & 31) << 2;          // float column (float4 granule)
        for (int it = 0; it < 16; ++it) {
            int lrow = it * 8 + r8;
            int j = row0 + lrow;
            if (j >= N_DST) j = N_DST - 1;       // clamp: garbage rows never stored
            const int* ci = child_idx + (long)j * LCH;
            float a0 = 0.f, a1 = 0.f, a2 = 0.f, a3 = 0.f;
            #pragma unroll
            for (int k = 0; k < NCHILD; ++k) {   // child 4 dropped by hc[:-1]
                int tok = token_ids[ci[k]];
                float4 v = *(const float4*)(emb + (long)tok * D + c4);
                a0 += v.x; a1 += v.y; a2 += v.z; a3 += v.w;
            }
            float* s = S + lrow * LSTRIDE + c4;
            s[0] = a0; s[1] = a1; s[2] = a2; s[3] = a3;
        }
    }
    __syncthreads();

    // ---- Phase 2: three 128x128 GEMMs on the matrix pipe ----
    v8f hI[8], hO[8], hU[8];

    load_w128_paired(W_ih, Wp, tid);
    __syncthreads();
    gemm_strip<8>(S, Wp, m0, ln16, hi, hI);
    __syncthreads();

    load_w128_paired(W_oh, Wp, tid);
    __syncthreads();
    gemm_strip<8>(S, Wp, m0, ln16, hi, hO);
    __syncthreads();

    load_w128_paired(W_uh, Wp, tid);
    __syncthreads();
    gemm_strip<8>(S, Wp, m0, ln16, hi, hU);

    // ---- Phase 3: gates + combine (child_c == 0 kills the forget branch) ----
    #pragma unroll
    for (int t = 0; t < 8; ++t) {
        int n = t * 16 + ln16;
        float bi = b_ih[n], bo = b_oh[n], bu = b_uh[n];
        #pragma unroll
        for (int r = 0; r < 8; ++r) {
            float iv = sigmoid_f(hI[t][r] + bi);
            float ov = sigmoid_f(hO[t][r] + bo);
            float uv = tanhf(hU[t][r] + bu);
            float cv = iv * uv;                  // + (f*child_c).sum == 0
            hU[t][r] = ov * tanhf(cv);
        }
    }

    // ---- Phase 4: LayerNorm over the 128 columns of each row ----
    // Row (m0 + r + 8*hi) lives in vector slot r across all 8 tiles of one half-wave.
    float mean[8], rstd[8];
    #pragma unroll
    for (int r = 0; r < 8; ++r) {
        float s1 = 0.f, s2 = 0.f;
        #pragma unroll
        for (int t = 0; t < 8; ++t) { float v = hU[t][r]; s1 += v; s2 += v * v; }
        s1 += __shfl_xor(s1, 1); s1 += __shfl_xor(s1, 2);
        s1 += __shfl_xor(s1, 4); s1 += __shfl_xor(s1, 8);   // stays in 16-lane half
        s2 += __shfl_xor(s2, 1); s2 += __shfl_xor(s2, 2);
        s2 += __shfl_xor(s2, 4); s2 += __shfl_xor(s2, 8);
        float mu  = s1 * (1.0f / 128.0f);
        float var = s2 * (1.0f / 128.0f) - mu * mu;
        mean[r] = mu;
        rstd[r] = rsqrtf(var + 1e-5f);
    }

    __syncthreads();   // all waves done reading S before overwriting with h_norm

    #pragma unroll
    for (int t = 0; t < 8; ++t) {
        int n = t * 16 + ln16;
        float g = ln_g[n], bb = ln_b[n];
        #pragma unroll
        for (int r = 0; r < 8; ++r) {
            int m = m0 + r + 8 * hi;
            S[m * LSTRIDE + n] = (hU[t][r] - mean[r]) * rstd[r] * g + bb;
        }
    }

    // ---- Phase 5: classifier GEMM 128x104 (padded to 112 = 7 N-tiles) ----
    {
        // stage W_fc (128x104) zero-padded to 112 cols, K-paired layout
        for (int it = 0; it < 28; ++it) {
            int idx = tid + it * THREADS;        // 0..7167: (k2, col)
            int k2  = idx / NCLS_PAD;
            int col = idx - k2 * NCLS_PAD;
            float x0 = (col < NCLS) ? W_fc[(2 * k2)     * NCLS + col] : 0.0f;
            float x1 = (col < NCLS) ? W_fc[(2 * k2 + 1) * NCLS + col] : 0.0f;
            v2f p = {x0, x1};
            *(v2f*)(Wp + k2 * WSTRIDE2 + 2 * col) = p;
        }
    }
    __syncthreads();

    v8f accF[7];
    gemm_strip<7>(S, Wp, m0, ln16, hi, accF);

    #pragma unroll
    for (int t = 0; t < 7; ++t) {
        int n = t * 16 + ln16;
        float bf = (n < NCLS) ? b_fc[n] : 0.0f;
        #pragma unroll
        for (int r = 0; r < 8; ++r) {
            int j = row0 + m0 + r + 8 * hi;
            if (j < N_DST && n < NCLS)
                out[(long)j * NCLS + n] = accF[t][r] + bf;
        }
    }
}

extern "C" void kernel_launch(void* const* d_in, const int* in_sizes, int n_in,
                              void* d_out, int out_size, void* d_ws, size_t ws_size,
                              hipStream_t stream) {
    (void)in_sizes; (void)n_in; (void)out_size; (void)d_ws; (void)ws_size;
    const int*   token_ids = (const int*)d_in[0];
    const int*   child_idx = (const int*)d_in[1];
    const float* emb       = (const float*)d_in[2];
    const float* W_ih      = (const float*)d_in[3];
    const float* b_ih      = (const float*)d_in[4];
    const float* W_oh      = (const float*)d_in[5];
    const float* b_oh      = (const float*)d_in[6];
    const float* W_uh      = (const float*)d_in[7];
    const float* b_uh      = (const float*)d_in[8];
    // d_in[9] = W_fh, d_in[10] = b_fh: unused — child_c is identically zero.
    const float* ln_g      = (const float*)d_in[11];
    const float* ln_b      = (const float*)d_in[12];
    const float* W_fc      = (const float*)d_in[13];
    const float* b_fc      = (const float*)d_in[14];
    float* out = (float*)d_out;

    int grid = (N_DST + BLOCK_ROWS - 1) / BLOCK_ROWS;   // 782
    size_t shmem = (size_t)(BLOCK_ROWS * LSTRIDE + 64 * WSTRIDE2) * sizeof(float); // 141312 B
    treelstm_cls_kernel<<<grid, THREADS, shmem, stream>>>(
        token_ids, child_idx, emb,
        W_ih, b_ih, W_oh, b_oh, W_uh, b_uh,
        ln_g, ln_b, W_fc, b_fc, out);
}